// LureSystem_47021301956813
// MI455X (gfx1250) — compile-verified
//
#include <hip/hip_runtime.h>

typedef __attribute__((ext_vector_type(2))) float v2f;
typedef __attribute__((ext_vector_type(8))) float v8f;

namespace {
constexpr int kNX = 64;    // state dim
constexpr int kND = 16;    // input dim
constexpr int kNE = 16;    // output dim
constexpr int kNW = 64;    // nonlinearity dim
constexpr int kB  = 256;   // batch
constexpr int kN  = 1024;  // time steps
constexpr int kNB = 16;    // batch columns per workgroup == WMMA N dim
constexpr int kXS = 68;    // LDS row stride (floats) for 64-wide transposed tiles
constexpr int kDS = 20;    // LDS row stride (floats) for 16-wide transposed tiles
}

// D = A(16x4 f32) * B(4x16 f32) + C  -- full-precision CDNA5 WMMA
static __device__ __forceinline__ v8f wmma4(v2f a, v2f b, v8f c) {
  return __builtin_amdgcn_wmma_f32_16x16x4_f32(
      /*neg_a=*/false, a, /*neg_b=*/false, b,
      /*c_mod=*/(short)0, c, /*reuse_a=*/false, /*reuse_b=*/false);
}

static __device__ __forceinline__ float tanh_stable(float x) {
  // 1 - 2/(e^{2x}+1): saturates to +/-1 without inf/inf NaNs
  float e2 = __expf(2.0f * x);
  return 1.0f - 2.0f / (e2 + 1.0f);
}

// One workgroup handles a 16-batch tile for all 1024 steps.
// Waves 0..3: rows 16w..16w+15 of W (phase 1) and x_next (phase 2).
// Wave 4: prefetch next d slice (phase 1) and compute E (phase 2).
__global__ __launch_bounds__(160)
void lure_rollout_wmma(const float* __restrict__ d,
                       const float* __restrict__ x0,
                       const float* __restrict__ A,
                       const float* __restrict__ B,
                       const float* __restrict__ C,
                       const float* __restrict__ Dm,
                       const float* __restrict__ B2,
                       const float* __restrict__ C2,
                       const float* __restrict__ D12,
                       const float* __restrict__ D21,
                       float* __restrict__ out_e,
                       float* __restrict__ out_x,
                       float* __restrict__ out_w)
{
  __shared__ float Xbuf[2][kNB * kXS];  // state, transposed [n][m], double buffered
  __shared__ float Wbuf[kNB * kXS];     // tanh output, transposed [n][m]
  __shared__ float Dbuf[2][kNB * kDS];  // d slice, transposed [n][c], double buffered

  const int tid  = threadIdx.x;
  const int wave = tid >> 5;
  const int lane = tid & 31;
  const int lmod = lane & 15;   // N column (batch within tile) / A-matrix row offset
  const int lh   = lane >> 4;   // lane half: selects K pair / M half
  const int b0   = blockIdx.x * kNB;

  // ---- setup: x0 -> Xbuf[0] (transposed) and x[:,0,:] output; d step 0 -> Dbuf[0]
  for (int i = tid; i < kNB * kNX; i += blockDim.x) {
    int n = i >> 6, c = i & 63;
    float v = x0[(size_t)(b0 + n) * kNX + c];
    Xbuf[0][n * kXS + c] = v;
    out_x[(size_t)(b0 + n) * (kN + 1) * kNX + c] = v;
  }
  for (int i = tid; i < kNB * kND; i += blockDim.x) {
    int n = i >> 4, c = i & 15;
    Dbuf[0][n * kDS + c] = d[((size_t)(b0 + n) * kN) * kND + c];
  }

  if (wave < 4) {
    // ================= compute waves =================
    const int rb = wave * 16;  // row base of this wave's 16x16 output tiles

    // Preload all weight A-fragments into registers (persist across 1024 steps).
    // A-frag j (K window 4j..4j+3): lane holds {M[rb+lmod][4j+2lh], M[rb+lmod][4j+2lh+1]}
    v2f fC2[16], fA[16], fB2[16], fD21[4], fB[4];
#pragma unroll
    for (int j = 0; j < 16; ++j) {
      int kk = 4 * j + 2 * lh;
      fC2[j] = *(const v2f*)&C2[(rb + lmod) * kNX + kk];
      fA [j] = *(const v2f*)&A [(rb + lmod) * kNX + kk];
      fB2[j] = *(const v2f*)&B2[(rb + lmod) * kNW + kk];
    }
#pragma unroll
    for (int j = 0; j < 4; ++j) {
      int kk = 4 * j + 2 * lh;
      fD21[j] = *(const v2f*)&D21[(rb + lmod) * kND + kk];
      fB  [j] = *(const v2f*)&B  [(rb + lmod) * kND + kk];
    }

    float* wout = out_w + (size_t)(b0 + lmod) * kN * kNW + (rb + 8 * lh);
    float* xout = out_x + (size_t)(b0 + lmod) * (kN + 1) * kNX + (rb + 8 * lh);

    __syncthreads();

    for (int k = 0; k < kN; ++k) {
      const int cur = k & 1, nxt = cur ^ 1;
      const float* Xc = &Xbuf[cur][lmod * kXS + 2 * lh];
      const float* Dc = &Dbuf[cur][lmod * kDS + 2 * lh];

      // ---- phase 1: W[rb..rb+15] = tanh(C2*X + D21*Dk) ----
      v8f a0 = {}, a1 = {};
#pragma unroll
      for (int j = 0; j < 16; j += 2) {
        v2f bf0 = *(const v2f*)&Xc[4 * j];
        v2f bf1 = *(const v2f*)&Xc[4 * (j + 1)];
        a0 = wmma4(fC2[j],     bf0, a0);
        a1 = wmma4(fC2[j + 1], bf1, a1);
      }
#pragma unroll
      for (int j = 0; j < 4; j += 2) {
        v2f bf0 = *(const v2f*)&Dc[4 * j];
        v2f bf1 = *(const v2f*)&Dc[4 * (j + 1)];
        a0 = wmma4(fD21[j],     bf0, a0);
        a1 = wmma4(fD21[j + 1], bf1, a1);
      }
      v8f wacc = a0 + a1;
#pragma unroll
      for (int v = 0; v < 8; ++v) wacc[v] = tanh_stable(wacc[v]);

      {  // store W tile: LDS transposed (for phase 2 B-operands) + global
        float* wl = &Wbuf[lmod * kXS + rb + 8 * lh];
        float* wg = wout + (size_t)k * kNW;
#pragma unroll
        for (int v = 0; v < 8; ++v) { wl[v] = wacc[v]; wg[v] = wacc[v]; }
      }
      __syncthreads();

      // ---- phase 2: x_next[rb..rb+15] = A*X + B*Dk + B2*W ----
      v8f c0 = {}, c1 = {};
#pragma unroll
      for (int j = 0; j < 16; j += 2) {
        v2f bf0 = *(const v2f*)&Xc[4 * j];
        v2f bf1 = *(const v2f*)&Xc[4 * (j + 1)];
        c0 = wmma4(fA[j],     bf0, c0);
        c1 = wmma4(fA[j + 1], bf1, c1);
      }
#pragma unroll
      for (int j = 0; j < 4; j += 2) {
        v2f bf0 = *(const v2f*)&Dc[4 * j];
        v2f bf1 = *(const v2f*)&Dc[4 * (j + 1)];
        c0 = wmma4(fB[j],     bf0, c0);
        c1 = wmma4(fB[j + 1], bf1, c1);
      }
      const float* Wc = &Wbuf[lmod * kXS + 2 * lh];
#pragma unroll
      for (int j = 0; j < 16; j += 2) {
        v2f bf0 = *(const v2f*)&Wc[4 * j];
        v2f bf1 = *(const v2f*)&Wc[4 * (j + 1)];
        c0 = wmma4(fB2[j],     bf0, c0);
        c1 = wmma4(fB2[j + 1], bf1, c1);
      }
      v8f xacc = c0 + c1;
      {  // store x_next: LDS transposed (next step's state) + global
        float* xl = &Xbuf[nxt][lmod * kXS + rb + 8 * lh];
        float* xg = xout + (size_t)(k + 1) * kNX;
#pragma unroll
        for (int v = 0; v < 8; ++v) { xl[v] = xacc[v]; xg[v] = xacc[v]; }
      }
      __syncthreads();
    }
  } else {
    // ================= wave 4: d prefetch + E tile =================
    v2f fC[16], fD12[16], fD[4];
#pragma unroll
    for (int j = 0; j < 16; ++j) {
      int kk = 4 * j + 2 * lh;
      fC  [j] = *(const v2f*)&C  [lmod * kNX + kk];
      fD12[j] = *(const v2f*)&D12[lmod * kNW + kk];
    }
#pragma unroll
    for (int j = 0; j < 4; ++j) {
      int kk = 4 * j + 2 * lh;
      fD[j] = *(const v2f*)&Dm[lmod * kND + kk];
    }

    float* eout = out_e + (size_t)(b0 + lmod) * kN * kNE + 8 * lh;

    __syncthreads();

    for (int k = 0; k < kN; ++k) {
      const int cur = k & 1, nxt = cur ^ 1;

      // phase 1 slot: stage next step's d slice into the other buffer
      if (k + 1 < kN) {
        for (int i = lane; i < kNB * kND; i += 32) {
          int n = i >> 4, c = i & 15;
          Dbuf[nxt][n * kDS + c] =
              d[((size_t)(b0 + n) * kN + (k + 1)) * kND + c];
        }
      }
      if (k + 8 < kN)  // warm the d stream ahead (global_prefetch_b8)
        __builtin_prefetch(&d[((size_t)(b0 + lmod) * kN + (k + 8)) * kND], 0, 1);
      __syncthreads();

      // phase 2 slot: E = C*X + D*Dk + D12*W
      const float* Xc = &Xbuf[cur][lmod * kXS + 2 * lh];
      const float* Dc = &Dbuf[cur][lmod * kDS + 2 * lh];
      const float* Wc = &Wbuf[lmod * kXS + 2 * lh];
      v8f e0 = {}, e1 = {};
#pragma unroll
      for (int j = 0; j < 16; j += 2) {
        v2f bf0 = *(const v2f*)&Xc[4 * j];
        v2f bf1 = *(const v2f*)&Xc[4 * (j + 1)];
        e0 = wmma4(fC[j],     bf0, e0);
        e1 = wmma4(fC[j + 1], bf1, e1);
      }
#pragma unroll
      for (int j = 0; j < 4; j += 2) {
        v2f bf0 = *(const v2f*)&Dc[4 * j];
        v2f bf1 = *(const v2f*)&Dc[4 * (j + 1)];
        e0 = wmma4(fD[j],     bf0, e0);
        e1 = wmma4(fD[j + 1], bf1, e1);
      }
#pragma unroll
      for (int j = 0; j < 16; j += 2) {
        v2f bf0 = *(const v2f*)&Wc[4 * j];
        v2f bf1 = *(const v2f*)&Wc[4 * (j + 1)];
        e0 = wmma4(fD12[j],     bf0, e0);
        e1 = wmma4(fD12[j + 1], bf1, e1);
      }
      v8f eacc = e0 + e1;
      {
        float* eg = eout + (size_t)k * kNE;
#pragma unroll
        for (int v = 0; v < 8; ++v) eg[v] = eacc[v];
      }
      __syncthreads();
    }
  }
}

extern "C" void kernel_launch(void* const* d_in, const int* in_sizes, int n_in,
                              void* d_out, int out_size, void* d_ws, size_t ws_size,
                              hipStream_t stream) {
  (void)in_sizes; (void)n_in; (void)out_size; (void)d_ws; (void)ws_size;
  const float* d   = (const float*)d_in[0];
  const float* x0  = (const float*)d_in[1];
  const float* A   = (const float*)d_in[2];
  const float* B   = (const float*)d_in[3];
  const float* C   = (const float*)d_in[4];
  const float* Dm  = (const float*)d_in[5];
  const float* B2  = (const float*)d_in[6];
  const float* C2  = (const float*)d_in[7];
  const float* D12 = (const float*)d_in[8];
  const float* D21 = (const float*)d_in[9];

  float* out = (float*)d_out;
  const size_t e_elems = (size_t)kB * kN * kNE;         //  4,194,304
  const size_t x_elems = (size_t)kB * (kN + 1) * kNX;   // 16,793,600
  const size_t w_elems = (size_t)kB * kN * kNW;         // 16,777,216
  const size_t d_elems = (size_t)kB * kN * kND;         //  4,194,304
  float* out_e = out;
  float* out_x = out_e + e_elems;
  float* out_w = out_x + x_elems;
  float* out_d = out_w + w_elems;

  // passthrough output: d
  hipMemcpyAsync(out_d, d, d_elems * sizeof(float), hipMemcpyDeviceToDevice, stream);

  dim3 grid(kB / kNB);   // 16 workgroups (one 16-batch tile each)
  dim3 block(160);       // 5 wave32's
  lure_rollout_wmma<<<grid, block, 0, stream>>>(d, x0, A, B, C, Dm, B2, C2, D12, D21,
                                                out_e, out_x, out_w);
}